// LSSTransformer_45818711113764
// MI455X (gfx1250) — compile-verified
//
#include <hip/hip_runtime.h>
#include <math.h>

typedef float v2f __attribute__((ext_vector_type(2)));
typedef float v8f __attribute__((ext_vector_type(8)));

#define NUM_VIEWS 6
#define DEPTH     60
#define NUM_POINTS 10
#define GRID_H    192
#define GRID_W    192
#define FEAT_H    48
#define FEAT_W    120
#define B_        2
#define C_IN      128
#define NPIX      (FEAT_H * FEAT_W)       // 5760
#define NCAM      (B_ * NUM_VIEWS)        // 12
#define DPAD      64                      // depth padded 60 -> 64
#define NCELLS    (B_ * GRID_H * GRID_W)  // 73728

// ---------------------------------------------------------------------------
// Kernel 1: per-pixel GEMMs via V_WMMA_F32_16X16X4_F32 (full f32 precision),
// depth softmax fused. One wave handles 16 pixels of one camera.
//   D = A x B + C with  A: 16(out-ch) x 4(K)  weights
//                       B: 4(K) x 16(pixels)  feats^T
//   K = 128 -> 32 chained WMMA per 16x16 tile; 4 depth tiles + 8 feat tiles.
// Outputs: dprob [12][5760][64] (pixel-major), featp [12][5760][128].
// ---------------------------------------------------------------------------
__global__ __launch_bounds__(128) void k1_gemm_softmax(
    const float* __restrict__ feats,    // (12,128,48,120)
    const float* __restrict__ w_depth,  // (60,128)
    const float* __restrict__ w_feat,   // (128,128)
    float* __restrict__ dprob,          // (12,5760,64)
    float* __restrict__ featp)          // (12,5760,128)
{
    const int lane = threadIdx.x & 31;
    const int wave = blockIdx.x * (blockDim.x >> 5) + (threadIdx.x >> 5);
    const int n    = wave / (NPIX / 16);        // camera 0..11
    const int pg   = wave % (NPIX / 16);        // pixel group
    const int lo   = lane & 15;                 // N=pixel idx / A row idx
    const int hi   = lane >> 4;                 // lane half -> K pair / M half
    const int pix  = pg * 16 + lo;

    // ---- preload B fragments (feats^T: K=128 x N=16) once, reuse x12 tiles.
    // step s covers channels 4s..4s+3; this half holds {4s+2hi, 4s+2hi+1}.
    const float* fbase = feats + (size_t)n * C_IN * NPIX + pix;
    v2f bfrag[32];
#pragma unroll
    for (int s = 0; s < 32; ++s) {
        const int ch = 4 * s + 2 * hi;
        bfrag[s].x = fbase[(size_t)ch * NPIX];
        bfrag[s].y = fbase[(size_t)(ch + 1) * NPIX];
    }

    // ---- depth logits: 4 tiles of 16 rows (rows 60..63 are zero-padded)
    v8f accd[4];
#pragma unroll
    for (int t = 0; t < 4; ++t) {
        v8f acc = {};
        const int  row   = t * 16 + lo;
        const bool valid = (row < DEPTH);           // folds to true for t<3
        const float* wrow = w_depth + row * C_IN + 2 * hi;
#pragma unroll
        for (int s = 0; s < 32; ++s) {
            v2f a;
            a.x = valid ? wrow[4 * s]     : 0.0f;
            a.y = valid ? wrow[4 * s + 1] : 0.0f;
            acc = __builtin_amdgcn_wmma_f32_16x16x4_f32(
                false, a, false, bfrag[s], (short)0, acc, false, false);
        }
        accd[t] = acc;
    }

    // ---- softmax over depth. Element (t,j) in this lane is depth
    // d = t*16 + hi*8 + j of pixel `lo`; partner lane is xor 16.
    float m = -INFINITY;
#pragma unroll
    for (int t = 0; t < 4; ++t)
#pragma unroll
        for (int j = 0; j < 8; ++j) {
            const int d = t * 16 + hi * 8 + j;
            const float v = (d < DEPTH) ? accd[t][j] : -INFINITY;
            m = fmaxf(m, v);
        }
    m = fmaxf(m, __shfl_xor(m, 16, 32));

    float ssum = 0.0f;
#pragma unroll
    for (int t = 0; t < 4; ++t)
#pragma unroll
        for (int j = 0; j < 8; ++j) {
            const int d = t * 16 + hi * 8 + j;
            const float e = (d < DEPTH) ? __expf(accd[t][j] - m) : 0.0f;
            accd[t][j] = e;
            ssum += e;
        }
    ssum += __shfl_xor(ssum, 16, 32);
    const float inv = 1.0f / ssum;

    float* dout = dprob + ((size_t)n * NPIX + pix) * DPAD;
#pragma unroll
    for (int t = 0; t < 4; ++t) {
        const int dbase = t * 16 + hi * 8;
        *(float4*)(dout + dbase) =
            make_float4(accd[t][0] * inv, accd[t][1] * inv,
                        accd[t][2] * inv, accd[t][3] * inv);
        *(float4*)(dout + dbase + 4) =
            make_float4(accd[t][4] * inv, accd[t][5] * inv,
                        accd[t][6] * inv, accd[t][7] * inv);
    }

    // ---- feature projection: 8 tiles of 16 output channels
    float* fout = featp + ((size_t)n * NPIX + pix) * C_IN;
#pragma unroll
    for (int t = 0; t < 8; ++t) {
        v8f acc = {};
        const float* wrow = w_feat + (t * 16 + lo) * C_IN + 2 * hi;
#pragma unroll
        for (int s = 0; s < 32; ++s) {
            v2f a;
            a.x = wrow[4 * s];
            a.y = wrow[4 * s + 1];
            acc = __builtin_amdgcn_wmma_f32_16x16x4_f32(
                false, a, false, bfrag[s], (short)0, acc, false, false);
        }
        const int cbase = t * 16 + hi * 8;
        *(float4*)(fout + cbase)     = make_float4(acc[0], acc[1], acc[2], acc[3]);
        *(float4*)(fout + cbase + 4) = make_float4(acc[4], acc[5], acc[6], acc[7]);
    }
}

// ---------------------------------------------------------------------------
// Kernel 2: LSS geometry. One thread per BEV cell: project 20 z x 6 views,
// compute ranks exactly like the reference, keep the 10 smallest (stable).
// Equal ranks decode to identical gathers, so scan-order ties are harmless.
// ---------------------------------------------------------------------------
__global__ __launch_bounds__(256) void k2_geometry(
    const float* __restrict__ ego2img,  // (12,4,4)
    int* __restrict__ ranks)            // (NCELLS,10)
{
    const int cell = blockIdx.x * blockDim.x + threadIdx.x;
    if (cell >= NCELLS) return;
    const int w = cell % GRID_W;
    const int h = (cell / GRID_W) % GRID_H;
    const int b = cell / (GRID_W * GRID_H);

    const float res = 102.4f / GRID_W;
    const float mn  = -102.4f * 0.5f + res * 0.5f;
    const float xw  = mn + w * res;
    const float yw  = mn + h * res;
    const float sx  = (float)FEAT_W / 1920.0f;
    const float sy  = (float)FEAT_H / 768.0f;

    int best[NUM_POINTS];
#pragma unroll
    for (int k = 0; k < NUM_POINTS; ++k) best[k] = 0x7fffffff;

    for (int v = 0; v < NUM_VIEWS; ++v) {
        const float* P = ego2img + (size_t)(b * NUM_VIEWS + v) * 16;
        const float p00 = P[0] * sx, p01 = P[1] * sx, p02 = P[2]  * sx, p03 = P[3] * sx;
        const float p10 = P[4] * sy, p11 = P[5] * sy, p12 = P[6]  * sy, p13 = P[7] * sy;
        const float p20 = P[8],      p21 = P[9],      p22 = P[10],      p23 = P[11];
        const float c0 = p00 * xw + p01 * yw + p03;
        const float c1 = p10 * xw + p11 * yw + p13;
        const float c2 = p20 * xw + p21 * yw + p23;
        for (int zi = 0; zi < 20; ++zi) {
            const float z  = -10.0f + (float)zi * (20.0f / 19.0f);
            const float u  = c0 + p02 * z;
            const float vv = c1 + p12 * z;
            const float dd = c2 + p22 * z;
            const float dc = fmaxf(dd, 0.05f);
            int X = (int)(u / dc);                 // trunc-to-zero, like jnp cast
            int Y = (int)(vv / dc);
            int D = (int)dd;
            const bool invalid = (X < 0) | (X >= FEAT_W) | (Y < 0) |
                                 (Y >= FEAT_H) | (D < 0) | (D >= DEPTH);
            if (invalid) { X = FEAT_W - 1; Y = FEAT_H - 1; D = DEPTH; }
            const int iv = invalid ? (NUM_VIEWS - 1) : v;
            const int r  = ((D * FEAT_H + Y) * FEAT_W + X) * NUM_VIEWS + iv;
            if (r < best[NUM_POINTS - 1]) {
                best[NUM_POINTS - 1] = r;
#pragma unroll
                for (int k = NUM_POINTS - 2; k >= 0; --k) {
                    if (best[k] > best[k + 1]) {
                        const int t = best[k]; best[k] = best[k + 1]; best[k + 1] = t;
                    }
                }
            }
        }
    }

    int* rout = ranks + (size_t)cell * NUM_POINTS;
#pragma unroll
    for (int k = 0; k < NUM_POINTS; ++k) rout[k] = best[k];
}

// ---------------------------------------------------------------------------
// Kernel 3: gather + weighted splat. One wave per BEV cell; lane owns 4
// contiguous channels -> coalesced float4 gathers from featp (L2 resident).
// ---------------------------------------------------------------------------
__global__ __launch_bounds__(256) void k3_splat(
    const float* __restrict__ dprob,   // (12,5760,64)
    const float* __restrict__ featp,   // (12,5760,128)
    const int*  __restrict__ ranks,    // (NCELLS,10)
    float* __restrict__ out)           // (2,128,192,192)
{
    const int lane = threadIdx.x & 31;
    const int cell = blockIdx.x * (blockDim.x >> 5) + (threadIdx.x >> 5);
    if (cell >= NCELLS) return;
    const int w = cell % GRID_W;
    const int h = (cell / GRID_W) % GRID_H;
    const int b = cell / (GRID_W * GRID_H);

    float4 acc = make_float4(0.f, 0.f, 0.f, 0.f);
    const int* rk = ranks + (size_t)cell * NUM_POINTS;
#pragma unroll
    for (int k = 0; k < NUM_POINTS; ++k) {
        const int r = rk[k];
        const int D = r / (FEAT_H * FEAT_W * NUM_VIEWS);
        if (D >= DEPTH) continue;                  // invalid -> zero weight
        int rem = r % (FEAT_H * FEAT_W * NUM_VIEWS);
        const int Y  = rem / (FEAT_W * NUM_VIEWS);
        rem %= FEAT_W * NUM_VIEWS;
        const int X  = rem / NUM_VIEWS;
        const int iv = rem % NUM_VIEWS;
        const size_t pixoff = (size_t)(b * NUM_VIEWS + iv) * NPIX
                            + (size_t)(Y * FEAT_W + X);
        const float dw = dprob[pixoff * DPAD + D];
        const float4 f = *(const float4*)(featp + pixoff * C_IN + lane * 4);
        acc.x += dw * f.x; acc.y += dw * f.y;
        acc.z += dw * f.z; acc.w += dw * f.w;
    }

    const size_t hw = (size_t)GRID_H * GRID_W;
    const size_t obase = ((size_t)b * C_IN + lane * 4) * hw
                       + (size_t)h * GRID_W + w;
    out[obase]          = acc.x;
    out[obase + hw]     = acc.y;
    out[obase + 2 * hw] = acc.z;
    out[obase + 3 * hw] = acc.w;
}

// ---------------------------------------------------------------------------
extern "C" void kernel_launch(void* const* d_in, const int* in_sizes, int n_in,
                              void* d_out, int out_size, void* d_ws, size_t ws_size,
                              hipStream_t stream) {
    const float* feats   = (const float*)d_in[0];  // (12,128,48,120)
    const float* ego2img = (const float*)d_in[1];  // (12,4,4)
    const float* w_depth = (const float*)d_in[2];  // (60,128)
    const float* w_feat  = (const float*)d_in[3];  // (128,128)
    float* out = (float*)d_out;

    // workspace layout (~56 MB): dprob | featp | ranks
    char* ws = (char*)d_ws;
    float* dprob = (float*)ws;
    size_t off = (size_t)NCAM * NPIX * DPAD * sizeof(float);
    float* featp = (float*)(ws + off);
    off += (size_t)NCAM * NPIX * C_IN * sizeof(float);
    int* ranks = (int*)(ws + off);

    // K1: 12 cams * 360 pixel-groups = 4320 waves; 4 waves / block
    k1_gemm_softmax<<<(NCAM * (NPIX / 16)) / 4, 128, 0, stream>>>(
        feats, w_depth, w_feat, dprob, featp);

    // K2: one thread per BEV cell
    k2_geometry<<<(NCELLS + 255) / 256, 256, 0, stream>>>(ego2img, ranks);

    // K3: one wave per BEV cell; 8 waves / block
    k3_splat<<<NCELLS / 8, 256, 0, stream>>>(dprob, featp, ranks, out);
}